// GICPModel_35381940584853
// MI455X (gfx1250) — compile-verified
//
#include <hip/hip_runtime.h>

typedef float v2f __attribute__((ext_vector_type(2)));
typedef float v8f __attribute__((ext_vector_type(8)));

#define N_SRC 12000
#define N_TAR 12000
#define SRC_TILES (N_SRC / 16)     // 750
#define TAR_TILES (N_TAR / 16)     // 750

#define WAVES_PER_BLOCK 6          // 192 threads
#define CHUNK_PTS   192            // tar points staged per chunk (one per thread)
#define CHUNK_TILES (CHUNK_PTS / 16)                       // 12
#define N_CHUNKS    ((TAR_TILES + CHUNK_TILES - 1) / CHUNK_TILES)  // 63

static_assert(SRC_TILES % WAVES_PER_BLOCK == 0, "all blocks fully populated");

// ---------------------------------------------------------------------------
// CDNA5 async-copy + wait helpers (ASYNCcnt path). Inline asm per bridge doc.
// ---------------------------------------------------------------------------
typedef __attribute__((address_space(3))) void lds_void;

__device__ __forceinline__ unsigned lds_off(const void* p) {
    // generic -> LDS address space; AS3 pointers are 32-bit LDS byte offsets
    return (unsigned)(unsigned long long)(lds_void*)p;
}

__device__ __forceinline__ void async_copy_b128(const void* gsrc, void* ldst) {
    unsigned l = lds_off(ldst);
    asm volatile("global_load_async_to_lds_b128 %0, %1, off"
                 :: "v"(l), "v"(gsrc) : "memory");
}

__device__ __forceinline__ void wait_asynccnt0() {
#if defined(__has_builtin) && __has_builtin(__builtin_amdgcn_s_wait_asynccnt)
    __builtin_amdgcn_s_wait_asynccnt(0);
#else
    asm volatile("s_wait_asynccnt 0x0" ::: "memory");
#endif
}

// ---------------------------------------------------------------------------
// Kernel 1: fused K=4 GEMM + row argmin on V_WMMA_F32_16X16X4_F32.
// One wave owns 16 source points. Homogeneous-coordinate trick: tar.w == 1,
// so the K=3 cross term -2*tsrc[m,3]*tar[n,3] is row-constant and argmin-
// irrelevant; repurpose K=3 as A[m,3]=1, B[3,n]=|tar_n|^2 so that
//   D = A x B + 0 = |tar|^2 - 2*(tsrc.xyz . tar.xyz)      (C is inline zero)
// which is argmin-equivalent to the reference's full 4-D distance.
// tar tiles are double-buffer staged into LDS with global_load_async_to_lds.
// ---------------------------------------------------------------------------
__global__ __launch_bounds__(192) void gicp_nn_kernel(
    const float* __restrict__ T4,     // 4x4 row-major
    const float* __restrict__ src,    // [N_SRC,4]
    const float* __restrict__ tar,    // [N_TAR,4]
    int* __restrict__ outIdx)         // [N_SRC]
{
    __shared__ float4 tbuf[2][CHUNK_PTS];      // 6 KB of the 320 KB WGP LDS

    const int tid    = threadIdx.x;
    const int waveId = tid >> 5;
    const int wave   = blockIdx.x * WAVES_PER_BLOCK + waveId;   // always < SRC_TILES
    const int lane   = tid & 31;
    const int half   = lane >> 4;              // 0: K=0,1 / rows M=v ; 1: K=2,3 / rows M=v+8
    const int l16    = lane & 15;

    const float4* src4 = (const float4*)src;
    const float4* tar4 = (const float4*)tar;

    // Transformation (uniform scalar loads)
    float Tm[16];
#pragma unroll
    for (int i = 0; i < 16; ++i) Tm[i] = T4[i];

    // A-matrix: lane holds row M=l16; half 0 -> (-2 t0, -2 t1), half 1 -> (-2 t2, 1)
    const float4 sp = src4[wave * 16 + l16];
    const float t0 = Tm[0]*sp.x + Tm[1]*sp.y + Tm[2]*sp.z  + Tm[3]*sp.w;
    const float t1 = Tm[4]*sp.x + Tm[5]*sp.y + Tm[6]*sp.z  + Tm[7]*sp.w;
    const float t2 = Tm[8]*sp.x + Tm[9]*sp.y + Tm[10]*sp.z + Tm[11]*sp.w;
    v2f A;
    A.x = half ? (-2.0f * t2) : (-2.0f * t0);
    A.y = half ? 1.0f         : (-2.0f * t1);

    const v8f Zc = {};                         // C = 0, loop-invariant

    float best[8];
    int   bidx[8];
#pragma unroll
    for (int v = 0; v < 8; ++v) { best[v] = 3.4028235e38f; bidx[v] = 0; }

    // ---- prime the async pipeline: stage chunk 0 into buffer 0 ----
    {
        const int p = 0 * CHUNK_PTS + tid;
        if (p < N_TAR) async_copy_b128(&tar4[p], &tbuf[0][tid]);
    }
    wait_asynccnt0();
    __syncthreads();

    for (int c = 0; c < N_CHUNKS; ++c) {
        const int cb = c & 1;

        // stage next chunk into the other buffer (overlapped with compute)
        if (c + 1 < N_CHUNKS) {
            const int p = (c + 1) * CHUNK_PTS + tid;
            if (p < N_TAR) async_copy_b128(&tar4[p], &tbuf[cb ^ 1][tid]);
        }

        const int tilesHere = (TAR_TILES - c * CHUNK_TILES) < CHUNK_TILES
                            ? (TAR_TILES - c * CHUNK_TILES) : CHUNK_TILES;

        for (int tt = 0; tt < tilesHere; ++tt) {
            const float4 tp = tbuf[cb][tt * 16 + l16];       // ds_load_b128 (broadcast)

            const float nrm = tp.x*tp.x + tp.y*tp.y + tp.z*tp.z + tp.w*tp.w;
            v2f B;                       // B 4x16: VGPR0 rows K=0/K=2, VGPR1 rows K=1/K=3
            B.x = half ? tp.z : tp.x;
            B.y = half ? nrm  : tp.y;

            // (neg_a, A, neg_b, B, c_mod, C, reuse_a, reuse_b)
            v8f D = __builtin_amdgcn_wmma_f32_16x16x4_f32(
                false, A, false, B, (short)0, Zc, false, false);

            const int col = c * CHUNK_PTS + tt * 16 + l16;
#pragma unroll
            for (int v = 0; v < 8; ++v) {
                const float d = D[v];
                if (d < best[v]) { best[v] = d; bidx[v] = col; }  // strict < keeps first min
            }
        }

        wait_asynccnt0();      // my async stores to LDS for chunk c+1 complete
        __syncthreads();       // all waves done reading chunk c; chunk c+1 visible
    }

    // Argmin across the 16 lanes of each half (xor masks < 16 never cross halves)
#pragma unroll
    for (int v = 0; v < 8; ++v) {
        float b  = best[v];
        int   bi = bidx[v];
#pragma unroll
        for (int off = 1; off < 16; off <<= 1) {
            const float ob = __shfl_xor(b,  off, 32);
            const int   oi = __shfl_xor(bi, off, 32);
            if (ob < b || (ob == b && oi < bi)) { b = ob; bi = oi; }
        }
        if (l16 == 0)
            outIdx[wave * 16 + v + half * 8] = bi;   // D slot v maps to row v + 8*half
    }
}

// ---------------------------------------------------------------------------
// Kernel 2: per-point Mahalanobis term + mean reduction.
// comb3 = (covT[idx] + T covS T^T)[:3,:3]; mah = r^T comb3^{-1} r (adjugate/det)
// r = tar[idx,:3] - (src @ T)[:3]
// ---------------------------------------------------------------------------
__global__ __launch_bounds__(256) void gicp_mah_kernel(
    const float* __restrict__ T4,
    const float* __restrict__ src,
    const float* __restrict__ tar,
    const float* __restrict__ covS,   // [N_SRC,4,4]
    const float* __restrict__ covT,   // [N_TAR,4,4]
    const int* __restrict__ idx,
    float* __restrict__ out)
{
    const int i = blockIdx.x * blockDim.x + threadIdx.x;
    float val = 0.0f;

    if (i < N_SRC) {
        float Tm[16];
#pragma unroll
        for (int k = 0; k < 16; ++k) Tm[k] = T4[k];

        const float4 sp = ((const float4*)src)[i];
        // pT = src @ T  (row-vector times matrix)
        const float p0 = sp.x*Tm[0] + sp.y*Tm[4] + sp.z*Tm[8]  + sp.w*Tm[12];
        const float p1 = sp.x*Tm[1] + sp.y*Tm[5] + sp.z*Tm[9]  + sp.w*Tm[13];
        const float p2 = sp.x*Tm[2] + sp.y*Tm[6] + sp.z*Tm[10] + sp.w*Tm[14];

        const int j = idx[i];
        const float4 tp = ((const float4*)tar)[j];
        const float r0 = tp.x - p0, r1 = tp.y - p1, r2 = tp.z - p2;

        // M = T * covS_i   (4x4)
        const float* Cs = covS + (size_t)i * 16;
        float M[16];
#pragma unroll
        for (int a = 0; a < 4; ++a)
#pragma unroll
            for (int b = 0; b < 4; ++b) {
                float s = 0.0f;
#pragma unroll
                for (int c = 0; c < 4; ++c) s += Tm[a*4+c] * Cs[c*4+b];
                M[a*4+b] = s;
            }

        // comb3[a][l] = covT[j][a][l] + sum_b M[a][b] * T[l][b]   (a,l in 0..2)
        const float* Ct = covT + (size_t)j * 16;
        float cmb[9];
#pragma unroll
        for (int a = 0; a < 3; ++a)
#pragma unroll
            for (int l = 0; l < 3; ++l) {
                float s = Ct[a*4+l];
#pragma unroll
                for (int b = 0; b < 4; ++b) s += M[a*4+b] * Tm[l*4+b];
                cmb[a*3+l] = s;
            }

        const float c00 = cmb[0], c01 = cmb[1], c02 = cmb[2];
        const float c10 = cmb[3], c11 = cmb[4], c12 = cmb[5];
        const float c20 = cmb[6], c21 = cmb[7], c22 = cmb[8];

        // adjugate
        const float a00 = c11*c22 - c12*c21;
        const float a01 = c02*c21 - c01*c22;
        const float a02 = c01*c12 - c02*c11;
        const float a10 = c12*c20 - c10*c22;
        const float a11 = c00*c22 - c02*c20;
        const float a12 = c02*c10 - c00*c12;
        const float a20 = c10*c21 - c11*c20;
        const float a21 = c01*c20 - c00*c21;
        const float a22 = c00*c11 - c01*c10;
        const float det = c00*a00 + c01*a10 + c02*a20;

        const float q = r0*(a00*r0 + a01*r1 + a02*r2)
                      + r1*(a10*r0 + a11*r1 + a12*r2)
                      + r2*(a20*r0 + a21*r1 + a22*r2);
        val = q / det;
    }

    // wave32 shuffle reduction, then cross-wave via LDS
#pragma unroll
    for (int off = 16; off > 0; off >>= 1) val += __shfl_down(val, off, 32);

    __shared__ float ws[8];
    const int laneId = threadIdx.x & 31;
    const int waveId = threadIdx.x >> 5;
    if (laneId == 0) ws[waveId] = val;
    __syncthreads();

    if (threadIdx.x < 8) {
        float v = ws[threadIdx.x];
#pragma unroll
        for (int off = 4; off > 0; off >>= 1) v += __shfl_down(v, off, 32);
        if (threadIdx.x == 0) {
            const float scaled = v * (1.0f / (float)N_SRC);
            __hip_atomic_fetch_add(out, scaled, __ATOMIC_RELAXED, __HIP_MEMORY_SCOPE_AGENT);
        }
    }
}

__global__ void gicp_zero_kernel(float* out) {
    if (threadIdx.x == 0 && blockIdx.x == 0) out[0] = 0.0f;
}

extern "C" void kernel_launch(void* const* d_in, const int* in_sizes, int n_in,
                              void* d_out, int out_size, void* d_ws, size_t ws_size,
                              hipStream_t stream) {
    const float* T4   = (const float*)d_in[0];   // [4,4]
    const float* src  = (const float*)d_in[1];   // [N_SRC,4]
    const float* tar  = (const float*)d_in[2];   // [N_TAR,4]
    const float* covS = (const float*)d_in[3];   // [N_SRC,4,4]
    const float* covT = (const float*)d_in[4];   // [N_TAR,4,4]
    float* out = (float*)d_out;
    int* nnIdx = (int*)d_ws;                     // N_SRC ints of scratch

    gicp_zero_kernel<<<1, 32, 0, stream>>>(out);

    const int nnBlocks = SRC_TILES / WAVES_PER_BLOCK;   // 125 full blocks
    gicp_nn_kernel<<<nnBlocks, WAVES_PER_BLOCK * 32, 0, stream>>>(T4, src, tar, nnIdx);

    gicp_mah_kernel<<<(N_SRC + 255) / 256, 256, 0, stream>>>(
        T4, src, tar, covS, covT, nnIdx, out);
}